// A3TGCN_49400713839035
// MI455X (gfx1250) — compile-verified
//
#include <hip/hip_runtime.h>
#include <math.h>

typedef float v2f __attribute__((ext_vector_type(2)));
typedef float v8f __attribute__((ext_vector_type(8)));

#define F 32

#define WMMA_F32(A, B, C) \
  __builtin_amdgcn_wmma_f32_16x16x4_f32(false, (A), false, (B), (short)0, (C), false, false)

__device__ __forceinline__ float sigmoidf_(float x) { return 1.0f / (1.0f + expf(-x)); }

// Native device-scope fp32 atomic add (no return): guaranteed global_atomic_add_f32,
// RMW performed at L2 (device) scope. Tracked by STOREcnt; s_endpgm waits it out.
__device__ __forceinline__ void atomic_add_f32_dev(float* p, float v) {
  asm volatile("global_atomic_add_f32 %0, %1, off scope:SCOPE_DEV"
               :: "v"(p), "v"(v) : "memory");
}

// pair-interleaved weight packing: W[k][n] -> Wp[(k>>1)*64 + n*2 + (k&1)]
// so a lane's B fragment (two K-adjacent values, same n) is one aligned b64 load.
__device__ __forceinline__ int pk_idx(int k, int n) {
  return ((k >> 1) << 6) + (n << 1) + (k & 1);
}

// ---------------------------------------------------------------- utility
__global__ void k_init_deg(float* __restrict__ deg, int n) {
  int t = blockIdx.x * blockDim.x + threadIdx.x;
  if (t < n) deg[t] = 1.0f;  // self-loop weight 1.0
}

__global__ void k_scatter_deg(float* __restrict__ deg, const long long* __restrict__ dst,
                              const float* __restrict__ w, int E) {
  int e = blockIdx.x * blockDim.x + threadIdx.x;
  if (e >= E) return;
  atomic_add_f32_dev(&deg[dst[e]], w[e]);
}

__global__ void k_dinv(const float* __restrict__ deg, float* __restrict__ dinv, int n) {
  int t = blockIdx.x * blockDim.x + threadIdx.x;
  if (t >= n) return;
  float d = deg[t];
  dinv[t] = d > 0.0f ? rsqrtf(d) : 0.0f;
}

__global__ void k_norm(float* __restrict__ norm, const long long* __restrict__ src,
                       const long long* __restrict__ dst, const float* __restrict__ w,
                       const float* __restrict__ dinv, int E) {
  int e = blockIdx.x * blockDim.x + threadIdx.x;
  if (e >= E) return;
  norm[e] = dinv[src[e]] * w[e] * dinv[dst[e]];
}

__global__ void k_softmax(const float* __restrict__ att, float* __restrict__ probs, int P) {
  if (threadIdx.x == 0 && blockIdx.x == 0) {
    float m = att[0];
    for (int i = 1; i < P; ++i) m = fmaxf(m, att[i]);
    float s = 0.0f;
    for (int i = 0; i < P; ++i) s += expf(att[i] - m);
    float inv = 1.0f / s;
    for (int i = 0; i < P; ++i) probs[i] = expf(att[i] - m) * inv;
  }
}

// Fused: slice Xp[n,f] = X[n,f,p]  AND seed AXp with the self-loop term
// AXp[n,f] = dinv[n]^2 * Xp[n,f]   (replaces a zero pass + N atomic items)
__global__ void k_slice_init(const float* __restrict__ X, float* __restrict__ Xp,
                             float* __restrict__ AXp, const float* __restrict__ dinv,
                             int N, int P, int p) {
  int t = blockIdx.x * blockDim.x + threadIdx.x;
  if (t >= N * F) return;
  int n = t >> 5, f = t & 31;
  float x = X[(size_t)n * F * P + (size_t)f * P + p];
  float di = dinv[n];
  Xp[t] = x;
  AXp[t] = di * di * x;
}

// Fold weights (one block of 1024 threads), pair-interleaved packing:
//   slot 0..2 : Wz' = Wz@Wlz_top, Wr' = Wr@Wlr_top, Wh' = Wh@Wlh_top
//   slot 3    : Wlh_bottom     slot 4 : Wlz_bottom     slot 5 : Wlr_bottom
//   bpack     : b*' = bl* + b* @ Wl*_top
__global__ void k_fold(const float* __restrict__ Wz, const float* __restrict__ Wr,
                       const float* __restrict__ Wh, const float* __restrict__ Wlz,
                       const float* __restrict__ Wlr, const float* __restrict__ Wlh,
                       const float* __restrict__ bz, const float* __restrict__ br,
                       const float* __restrict__ bh, const float* __restrict__ blz,
                       const float* __restrict__ blr, const float* __restrict__ blh,
                       float* __restrict__ Wpack, float* __restrict__ bpack) {
  int t = threadIdx.x;           // 1024 threads: (k, n)
  int n = t & 31, k = t >> 5;
  float az = 0.0f, ar = 0.0f, ah = 0.0f;
  for (int j = 0; j < 32; ++j) {
    az += Wz[k * 32 + j] * Wlz[j * 32 + n];
    ar += Wr[k * 32 + j] * Wlr[j * 32 + n];
    ah += Wh[k * 32 + j] * Wlh[j * 32 + n];
  }
  int pi = pk_idx(k, n);
  Wpack[0 * 1024 + pi] = az;
  Wpack[1 * 1024 + pi] = ar;
  Wpack[2 * 1024 + pi] = ah;
  Wpack[3 * 1024 + pi] = Wlh[(32 + k) * 32 + n];
  Wpack[4 * 1024 + pi] = Wlz[(32 + k) * 32 + n];
  Wpack[5 * 1024 + pi] = Wlr[(32 + k) * 32 + n];
  if (k == 0) {
    float sz = blz[n], sr = blr[n], sh = blh[n];
    for (int j = 0; j < 32; ++j) {
      sz += bz[j] * Wlz[j * 32 + n];
      sr += br[j] * Wlr[j * 32 + n];
      sh += bh[j] * Wlh[j * 32 + n];
    }
    bpack[n] = sz; bpack[32 + n] = sr; bpack[64 + n] = sh;
  }
}

// ---------------------------------------------------------------- aggregation
// AXp[d,f] += norm[e] * Xp[s,f]   (E edges; self-loops already seeded).
// One lane per (edge, feature): 128B coalesced gather + native L2 atomic add.
__global__ void k_aggregate(const float* __restrict__ Xp, const long long* __restrict__ src,
                            const long long* __restrict__ dst, const float* __restrict__ norm,
                            float* __restrict__ AXp, int E) {
  long long t = (long long)blockIdx.x * blockDim.x + threadIdx.x;
  long long e = t >> 5;
  int f = (int)(t & 31);
  if (e >= E) return;
  long long s = src[e], d = dst[e];
  float v = norm[e] * Xp[s * F + f];
  atomic_add_f32_dev(&AXp[d * F + f], v);
}

// ---------------------------------------------------------------- WMMA GEMM
// Out(N x 32) = In(N x 32) @ W(32 x 32), W given pair-packed. 16-row tile per wave.
__global__ void k_gemm32(const float* __restrict__ A, const float* __restrict__ Wp,
                         float* __restrict__ Out, int N) {
  __shared__ float sW[32 * 32];
  for (int i = threadIdx.x; i < 1024; i += blockDim.x) sW[i] = Wp[i];
  __syncthreads();
  int wave = threadIdx.x >> 5, lane = threadIdx.x & 31;
  int tile = (blockIdx.x * (blockDim.x >> 5) + wave) * 16;
  if (tile >= N) return;                       // wave-uniform: EXEC stays all-1s
  const int r = lane & 15;
  const int kq = (lane >> 4) << 1;             // K sub-offset 0 or 2 (ISA 16x4 A layout)
  const int mb = (lane < 16) ? 0 : 8;          // C layout M base
  int rowA = tile + r; if (rowA > N - 1) rowA = N - 1;
  const float* Ar = A + (size_t)rowA * F;
  v8f c0 = {}, c1 = {};
  for (int k0 = 0; k0 < 32; k0 += 4) {
    v2f a = *(const v2f*)(Ar + k0 + kq);       // global_load_b64
    int pb = ((k0 + kq) >> 1) << 6;
    v2f b0 = *(const v2f*)(sW + pb + (r << 1));         // ds_load_b64
    v2f b1 = *(const v2f*)(sW + pb + ((16 + r) << 1));  // ds_load_b64
    c0 = WMMA_F32(a, b0, c0);
    c1 = WMMA_F32(a, b1, c1);
  }
  for (int v = 0; v < 8; ++v) {
    int row = tile + mb + v;
    if (row < N) {
      Out[(size_t)row * F + r]      = c0[v];
      Out[(size_t)row * F + 16 + r] = c1[v];
    }
  }
}

// ---------------------------------------------------------------- fused TGCN cell
// Z  = sigmoid(AXp@Wz' + HWz + bz')
// R  = sigmoid(AXp@Wr' + HWr + br')
// Ht = tanh   (AXp@Wh' + (H*R)@WlhB + bh')
// Hacc (+)= probs[p] * (Z*H + (1-Z)*Ht)
__global__ void k_cell(const float* __restrict__ AXp, const float* __restrict__ H,
                       const float* __restrict__ HWz, const float* __restrict__ HWr,
                       const float* __restrict__ Wpack, const float* __restrict__ bpack,
                       const float* __restrict__ probs, int p, int first,
                       float* __restrict__ Hacc, int N) {
  __shared__ float sW[4 * 1024];   // Wz', Wr', Wh', WlhB (pair-packed, 16 KB)
  __shared__ float sB[96];         // bz', br', bh'
  __shared__ float sHR[8][16 * F]; // per-wave C-layout -> A-layout bounce (16 KB)
  for (int i = threadIdx.x; i < 4 * 1024; i += blockDim.x) sW[i] = Wpack[i];
  for (int i = threadIdx.x; i < 96; i += blockDim.x) sB[i] = bpack[i];
  __syncthreads();

  int wave = threadIdx.x >> 5, lane = threadIdx.x & 31;
  int tile = (blockIdx.x * (blockDim.x >> 5) + wave) * 16;
  if (tile >= N) return;                       // wave-uniform exit
  const int r = lane & 15;
  const int kq = (lane >> 4) << 1;
  const int mb = (lane < 16) ? 0 : 8;
  const int c0i = r << 1, c1i = (16 + r) << 1;

  // --- 3 fused GEMMs over the shared A tile: 48 WMMAs ---
  int rowA = tile + r; if (rowA > N - 1) rowA = N - 1;
  const float* Ar = AXp + (size_t)rowA * F;
  const float* w0 = sW;
  const float* w1 = sW + 1024;
  const float* w2 = sW + 2048;
  v8f cz0 = {}, cz1 = {}, cr0 = {}, cr1 = {}, ch0 = {}, ch1 = {};
  for (int k0 = 0; k0 < 32; k0 += 4) {
    v2f a = *(const v2f*)(Ar + k0 + kq);
    int pb = ((k0 + kq) >> 1) << 6;
    v2f b;
    b = *(const v2f*)(w0 + pb + c0i); cz0 = WMMA_F32(a, b, cz0);
    b = *(const v2f*)(w0 + pb + c1i); cz1 = WMMA_F32(a, b, cz1);
    b = *(const v2f*)(w1 + pb + c0i); cr0 = WMMA_F32(a, b, cr0);
    b = *(const v2f*)(w1 + pb + c1i); cr1 = WMMA_F32(a, b, cr1);
    b = *(const v2f*)(w2 + pb + c0i); ch0 = WMMA_F32(a, b, ch0);
    b = *(const v2f*)(w2 + pb + c1i); ch1 = WMMA_F32(a, b, ch1);
  }

  // --- gates (C layout), build H*R tile in LDS ---
  float h0[8], h1[8], zl0[8], zl1[8];
  for (int v = 0; v < 8; ++v) {
    int row = tile + mb + v; if (row > N - 1) row = N - 1;
    size_t base = (size_t)row * F;
    h0[v] = H[base + r];  h1[v] = H[base + 16 + r];
    zl0[v] = sigmoidf_(cz0[v] + HWz[base + r]      + sB[r]);
    zl1[v] = sigmoidf_(cz1[v] + HWz[base + 16 + r] + sB[16 + r]);
    float R0 = sigmoidf_(cr0[v] + HWr[base + r]      + sB[32 + r]);
    float R1 = sigmoidf_(cr1[v] + HWr[base + 16 + r] + sB[32 + 16 + r]);
    sHR[wave][(mb + v) * F + r]      = h0[v] * R0;
    sHR[wave][(mb + v) * F + 16 + r] = h1[v] * R1;
  }
  // wave-local LDS RAW ordered via DScnt (compiler-inserted s_wait_dscnt)

  // --- 4th GEMM: (H*R) @ WlhB accumulated into ch: 16 WMMAs ---
  const float* w3 = sW + 3 * 1024;
  const float* AH = &sHR[wave][0];
  for (int k0 = 0; k0 < 32; k0 += 4) {
    v2f a = *(const v2f*)(AH + r * F + k0 + kq);   // ds_load_b64
    int pb = ((k0 + kq) >> 1) << 6;
    v2f b;
    b = *(const v2f*)(w3 + pb + c0i); ch0 = WMMA_F32(a, b, ch0);
    b = *(const v2f*)(w3 + pb + c1i); ch1 = WMMA_F32(a, b, ch1);
  }

  // --- combine + attention-weighted accumulate ---
  float pbv = probs[p];
  for (int v = 0; v < 8; ++v) {
    int row = tile + mb + v;
    if (row < N) {
      float ht0 = tanhf(ch0[v] + sB[64 + r]);
      float ht1 = tanhf(ch1[v] + sB[64 + 16 + r]);
      float hp0 = zl0[v] * h0[v] + (1.0f - zl0[v]) * ht0;
      float hp1 = zl1[v] * h1[v] + (1.0f - zl1[v]) * ht1;
      size_t base = (size_t)row * F;
      float o0 = first ? 0.0f : Hacc[base + r];
      float o1 = first ? 0.0f : Hacc[base + 16 + r];
      Hacc[base + r]      = o0 + pbv * hp0;
      Hacc[base + 16 + r] = o1 + pbv * hp1;
    }
  }
}

// ---------------------------------------------------------------- host
extern "C" void kernel_launch(void* const* d_in, const int* in_sizes, int n_in,
                              void* d_out, int out_size, void* d_ws, size_t ws_size,
                              hipStream_t stream) {
  const float*     X    = (const float*)d_in[0];
  const long long* ei   = (const long long*)d_in[1];   // int64 [2, E]
  const float*     ew   = (const float*)d_in[2];
  const float*     H    = (const float*)d_in[3];
  const float*     att  = (const float*)d_in[4];
  const float*     Wz   = (const float*)d_in[5];
  const float*     bz   = (const float*)d_in[6];
  const float*     Wr   = (const float*)d_in[7];
  const float*     br   = (const float*)d_in[8];
  const float*     Wh   = (const float*)d_in[9];
  const float*     bh   = (const float*)d_in[10];
  const float*     Wlz  = (const float*)d_in[11];
  const float*     blz  = (const float*)d_in[12];
  const float*     Wlr  = (const float*)d_in[13];
  const float*     blr  = (const float*)d_in[14];
  const float*     Wlh  = (const float*)d_in[15];
  const float*     blh  = (const float*)d_in[16];

  const int N = in_sizes[3] / F;
  const int E = in_sizes[2];
  const int P = in_sizes[4];
  const long long* srcIdx = ei;
  const long long* dstIdx = ei + E;

  // workspace carve-up (floats)
  float* ws    = (float*)d_ws;
  float* Xp    = ws;                ws += (size_t)N * F;
  float* AXp   = ws;                ws += (size_t)N * F;
  float* HWz   = ws;                ws += (size_t)N * F;
  float* HWr   = ws;                ws += (size_t)N * F;
  float* deg   = ws;                ws += N;
  float* dinv  = ws;                ws += N;
  float* norm  = ws;                ws += E;
  float* Wpack = ws;                ws += 6 * 1024;
  float* bpack = ws;                ws += 96;
  float* probs = ws;                ws += P;

  const int T = 256;
  const int nBlk    = (N + T - 1) / T;
  const int eBlk    = (E + T - 1) / T;
  const int nfBlk   = (N * F + T - 1) / T;
  const int tileBlk = (N + 16 * (T / 32) - 1) / (16 * (T / 32));
  long long aggThreads = (long long)E * 32;
  const int aggBlk = (int)((aggThreads + T - 1) / T);

  // graph normalization (once)
  k_init_deg   <<<nBlk, T, 0, stream>>>(deg, N);
  k_scatter_deg<<<eBlk, T, 0, stream>>>(deg, dstIdx, ew, E);
  k_dinv       <<<nBlk, T, 0, stream>>>(deg, dinv, N);
  k_norm       <<<eBlk, T, 0, stream>>>(norm, srcIdx, dstIdx, ew, dinv, E);

  // weight folding + packing, attention softmax (once)
  k_fold   <<<1, 1024, 0, stream>>>(Wz, Wr, Wh, Wlz, Wlr, Wlh,
                                    bz, br, bh, blz, blr, blh, Wpack, bpack);
  k_softmax<<<1, 32, 0, stream>>>(att, probs, P);

  // H @ Wl*_bot precomputed once (H identical across all periods)
  k_gemm32<<<tileBlk, T, 0, stream>>>(H, Wpack + 4 * 1024, HWz, N);
  k_gemm32<<<tileBlk, T, 0, stream>>>(H, Wpack + 5 * 1024, HWr, N);

  for (int p = 0; p < P; ++p) {
    k_slice_init<<<nfBlk, T, 0, stream>>>(X, Xp, AXp, dinv, N, P, p);
    k_aggregate <<<aggBlk, T, 0, stream>>>(Xp, srcIdx, dstIdx, norm, AXp, E);
    k_cell      <<<tileBlk, T, 0, stream>>>(AXp, H, HWz, HWr, Wpack, bpack,
                                            probs, p, p == 0 ? 1 : 0,
                                            (float*)d_out, N);
  }
}